// SampleModel_65017214927332
// MI455X (gfx1250) — compile-verified
//
#include <hip/hip_runtime.h>

typedef float v2f __attribute__((ext_vector_type(2)));
typedef float v8f __attribute__((ext_vector_type(8)));

#define D 256
#define TEMP_INV (1.0f / 0.07f)
#define NEPS 1e-12f
#define NPAD 260  // 16-col tile row stride in LDS dwords; 260%64=4 -> conflict-free

// ---------------- Kernel 1: L2-normalize centroid rows (wave per row) --------
__global__ void __launch_bounds__(256)
normalize_centroids(const float* __restrict__ c, float* __restrict__ cn, int K) {
  const int wave = threadIdx.x >> 5, lane = threadIdx.x & 31;
  const int r = blockIdx.x * 8 + wave;
  if (r >= K) return;
  const float4* src = reinterpret_cast<const float4*>(c + (size_t)r * D) + lane * 2;
  float4 x0 = src[0], x1 = src[1];
  float ss = x0.x * x0.x + x0.y * x0.y + x0.z * x0.z + x0.w * x0.w +
             x1.x * x1.x + x1.y * x1.y + x1.z * x1.z + x1.w * x1.w;
#pragma unroll
  for (int m = 16; m >= 1; m >>= 1) ss += __shfl_xor(ss, m, 32);
  const float s = 1.0f / fmaxf(sqrtf(ss), NEPS);
  float4* dst = reinterpret_cast<float4*>(cn + (size_t)r * D) + lane * 2;
  x0.x *= s; x0.y *= s; x0.z *= s; x0.w *= s;
  x1.x *= s; x1.y *= s; x1.z *= s; x1.w *= s;
  dst[0] = x0; dst[1] = x1;
}

// ---------------- Kernel 2: centsum[j] = sum_i exp(c_i . c_j / T) ------------
__global__ void __launch_bounds__(256)
centroid_partition(const float* __restrict__ cn, float* __restrict__ centsum, int K) {
  __shared__ float4 cj[D / 4];
  __shared__ float red[256];
  const int j = blockIdx.x, t = threadIdx.x;
  if (t < D / 4) cj[t] = reinterpret_cast<const float4*>(cn + (size_t)j * D)[t];
  __syncthreads();
  float acc = 0.0f;
  for (int i = t; i < K; i += 256) {
    const float4* ci = reinterpret_cast<const float4*>(cn + (size_t)i * D);
    float dot = 0.0f;
#pragma unroll
    for (int q = 0; q < D / 4; ++q) {
      float4 a = ci[q], b = cj[q];
      dot += a.x * b.x + a.y * b.y + a.z * b.z + a.w * b.w;
    }
    acc += expf(dot * TEMP_INV);
  }
  red[t] = acc;
  __syncthreads();
  for (int s2 = 128; s2 > 0; s2 >>= 1) {
    if (t < s2) red[t] += red[t + s2];
    __syncthreads();
  }
  if (t == 0) centsum[j] = red[0];
}

// Async global->LDS copy of 64 contiguous bytes (4 x b128), GV addressing.
// The shared inst offset is applied to BOTH the LDS and global addresses.
__device__ __forceinline__ void stage64_async(const float* gsrc, float* ldst) {
  const unsigned lds = (unsigned)(size_t)ldst;            // LDS byte offset
  const unsigned long long ga = (unsigned long long)gsrc; // 64-bit global addr
  asm volatile(
      "global_load_async_to_lds_b128 %0, %1, off\n\t"
      "global_load_async_to_lds_b128 %0, %1, off offset:16\n\t"
      "global_load_async_to_lds_b128 %0, %1, off offset:32\n\t"
      "global_load_async_to_lds_b128 %0, %1, off offset:48"
      :
      : "v"(lds), "v"(ga)
      : "memory");
}

__device__ __forceinline__ void wait_async0() {
  asm volatile("s_wait_asynccnt 0x0" ::: "memory");
}

// ---------------- Kernel 3: WMMA GEMM + fused row max/argmax + loss ----------
// One wave per 16-row feature tile; 8 waves/block share a double-buffered LDS
// centroid tile filled with GLOBAL_LOAD_ASYNC_TO_LDS_B128 (ASYNCcnt), so the
// staging latency of tile nt+1 overlaps the 64 chained v_wmma on tile nt.
__global__ void __launch_bounds__(256)
sim_max_loss(const float* __restrict__ feat, const float* __restrict__ cn,
             const float* __restrict__ centsum, float* __restrict__ partials,
             int Brows, int K) {
  __shared__ float bt[2][16 * NPAD];
  const int t = threadIdx.x;
  const int wave = t >> 5, lane = t & 31;
  const int mtile = blockIdx.x * 8 + wave;
  const int row0 = mtile * 16;
  const bool active = row0 < Brows;
  const int lrow = lane & 15;
  const int koff = (lane >> 4) * 2;

  // A fragments for all of K=256: 16x4 f32 layout (lanes0-15 K={0,1},
  // lanes16-31 K={2,3}), held in 64 v2f registers.
  v2f afrag[64];
  if (active) {
    int row = row0 + lrow;
    if (row >= Brows) row = Brows - 1;  // clamp (discarded later)
    const float* ap = feat + (size_t)row * D + koff;
#pragma unroll
    for (int kb = 0; kb < 64; ++kb)
      afrag[kb] = *reinterpret_cast<const v2f*>(ap + kb * 4);
  }

  float maxv[8];
  int maxi[8];
#pragma unroll
  for (int r = 0; r < 8; ++r) { maxv[r] = -3.4e38f; maxi[r] = 0; }

  const int ntiles = (K + 15) / 16;
  const int srow = t >> 4;        // staging: n within tile (0..15)
  const int sk = (t & 15) * 16;   // staging: 16-float chunk (coalesced)

  // Prologue: async-stage tile 0 into buffer 0.
  {
    int nrow = srow;
    if (nrow >= K) nrow = K - 1;
    stage64_async(cn + (size_t)nrow * D + sk, &bt[0][srow * NPAD + sk]);
  }
  wait_async0();
  __syncthreads();

  for (int nt = 0; nt < ntiles; ++nt) {
    const int n0 = nt * 16;
    const int cur = nt & 1;
    // Issue async staging of the NEXT tile into the other buffer first
    // (long-latency op overlaps the WMMA chain below).
    if (nt + 1 < ntiles) {
      int nrow = n0 + 16 + srow;
      if (nrow >= K) nrow = K - 1;
      stage64_async(cn + (size_t)nrow * D + sk,
                    &bt[cur ^ 1][srow * NPAD + sk]);
    }
    if (active) {
      v8f cacc = {0.f, 0.f, 0.f, 0.f, 0.f, 0.f, 0.f, 0.f};
      const float* bp = &bt[cur][lrow * NPAD + koff];
#pragma unroll
      for (int kb = 0; kb < 64; ++kb) {
        v2f b = *reinterpret_cast<const v2f*>(bp + kb * 4);
        cacc = __builtin_amdgcn_wmma_f32_16x16x4_f32(
            false, afrag[kb], false, b, (short)0, cacc, false, false);
      }
      const int col = n0 + lrow;
#pragma unroll
      for (int r = 0; r < 8; ++r) {
        const float v = cacc[r];
        if (col < K && v > maxv[r]) { maxv[r] = v; maxi[r] = col; }
      }
    }
    wait_async0();   // next tile's LDS writes complete (this wave's ops)
    __syncthreads(); // whole block done computing cur + staging next
  }

  if (active) {
    // butterfly (val,idx) reduce within each 16-lane half; first-index ties
#pragma unroll
    for (int r = 0; r < 8; ++r) {
      float v = maxv[r];
      int idx = maxi[r];
#pragma unroll
      for (int m = 8; m >= 1; m >>= 1) {
        const float ov = __shfl_xor(v, m, 32);
        const int oi = __shfl_xor(idx, m, 32);
        if (ov > v || (ov == v && oi < idx)) { v = ov; idx = oi; }
      }
      maxv[r] = v;
      maxi[r] = idx;
    }
    float acc = 0.0f;
    if (lrow == 0) {  // lane 0: rows 0-7, lane 16: rows 8-15
#pragma unroll
      for (int r = 0; r < 8; ++r) {
        const int row = row0 + r + (lane >> 4) * 8;
        if (row < Brows) {
          const float m = maxv[r] * TEMP_INV;
          const float s = centsum[maxi[r]];
          acc += log1pf(s * expf(-m));  // == -J_row
        }
      }
    }
    const float other = __shfl_down(acc, 16, 32);
    if (lane == 0) partials[mtile] = acc + other;
  }
}

// ---------------- Kernel 4: deterministic final reduction --------------------
__global__ void __launch_bounds__(256)
final_reduce(const float* __restrict__ partials, float* __restrict__ out,
             int mtiles, float invB) {
  __shared__ float red[256];
  const int t = threadIdx.x;
  float s = 0.0f;
  for (int i = t; i < mtiles; i += 256) s += partials[i];
  red[t] = s;
  __syncthreads();
  for (int k = 128; k > 0; k >>= 1) {
    if (t < k) red[t] += red[t + k];
    __syncthreads();
  }
  if (t == 0) out[0] = red[0] * invB;
}

extern "C" void kernel_launch(void* const* d_in, const int* in_sizes, int n_in,
                              void* d_out, int out_size, void* d_ws, size_t ws_size,
                              hipStream_t stream) {
  const float* feat = (const float*)d_in[0];
  const float* cent = (const float*)d_in[1];
  const int B = in_sizes[0] / D;  // 100000
  const int K = in_sizes[1] / D;  // 2000

  float* cn = (float*)d_ws;                     // K*D normalized centroids
  float* centsum = cn + (size_t)K * D;          // K partition sums
  const int mtiles = (B + 15) / 16;
  float* partials = centsum + K;                // mtiles per-tile loss partials
  float* out = (float*)d_out;

  normalize_centroids<<<(K + 7) / 8, 256, 0, stream>>>(cent, cn, K);
  centroid_partition<<<K, 256, 0, stream>>>(cn, centsum, K);
  sim_max_loss<<<(mtiles + 7) / 8, 256, 0, stream>>>(feat, cn, centsum,
                                                     partials, B, K);
  final_reduce<<<1, 256, 0, stream>>>(partials, out, mtiles, 1.0f / (float)B);
}